// PGExplainerModel_41437844472012
// MI455X (gfx1250) — compile-verified
//
#include <hip/hip_runtime.h>

typedef __attribute__((ext_vector_type(2))) float v2f;
typedef __attribute__((ext_vector_type(8))) float v8f;

#define HDIM 64

// ---------------------------------------------------------------------------
// Kernel 1: node-side precompute via fp32 WMMA.
//   P[n, j]      = sum_k emb[n,k] * W1[k, j]        (j in 0..63,  "top" block)
//   P[n, 64+j]   = sum_k emb[n,k] * W1[64+k, j]     (j in 0..63,  "bottom" block)
// i.e. one [N x 64] @ [64 x 128] GEMM into workspace P (row stride 128).
// Block = 256 threads = 8 waves; block b owns row-tile rows [16b, 16b+16),
// wave w owns column-tile w (16 columns). Each wave: 16 x V_WMMA_F32_16X16X4_F32.
// N % 16 == 0 (50000 = 3125*16) so EXEC is all-ones everywhere (WMMA req).
// ---------------------------------------------------------------------------
__global__ __launch_bounds__(256) void node_precompute_wmma(
    const float* __restrict__ emb,   // [N, 64]
    const float* __restrict__ W1,    // [128, 64] row-major
    float* __restrict__ P,           // [N, 128] workspace
    int N)
{
    const int ct   = threadIdx.x >> 5;   // column tile 0..7
    const int lane = threadIdx.x & 31;
    const int half = lane >> 4;          // 0: lanes 0-15, 1: lanes 16-31
    const int rc   = lane & 15;          // A: row-in-tile / B: col-in-tile
    const int rbase = blockIdx.x * 16;
    if (rbase >= N) return;              // uniform per block; EXEC stays full

    // A: 16x4 fp32 tile of emb. Lane layout (ISA 7.12.2):
    //   lanes 0-15: M=lane, VGPR0=K0, VGPR1=K1; lanes 16-31: M=lane-16, K2/K3.
    const float* arow = emb + (size_t)(rbase + rc) * HDIM;

    // B: 4x16 fp32 tile of W1' (mirrored layout: lanes 0-15 N=lane K0/K1,
    // lanes 16-31 N=lane-16 K2/K3). Column-tile ct<4 -> top block of W1,
    // ct>=4 -> bottom block (rows 64..127).
    const float* bcol = (ct < 4)
        ? (W1 + ct * 16 + rc)                       // element (k, j=ct*16+rc): +k*64
        : (W1 + 64 * 64 + (ct - 4) * 16 + rc);      // element (64+k, j): +k*64

    v8f c = {0.f, 0.f, 0.f, 0.f, 0.f, 0.f, 0.f, 0.f};

#pragma unroll
    for (int k0 = 0; k0 < HDIM; k0 += 4) {
        const int kb = k0 + half * 2;
        v2f a, b;
        a.x = arow[kb];
        a.y = arow[kb + 1];
        b.x = bcol[(size_t)kb * HDIM];
        b.y = bcol[(size_t)(kb + 1) * HDIM];
        // 8-arg form: (neg_a, A, neg_b, B, c_mod, C, reuse_a, reuse_b)
        c = __builtin_amdgcn_wmma_f32_16x16x4_f32(
                false, a, false, b, (short)0, c, false, false);
    }

    // C/D layout: VGPR v -> row (v + half*8), col rc (ISA 7.12.2).
    float* prow = P + (size_t)(rbase + half * 8) * 128 + ct * 16 + rc;
#pragma unroll
    for (int v = 0; v < 8; ++v)
        prow[(size_t)v * 128] = c[v];
}

// ---------------------------------------------------------------------------
// Kernel 2: per-edge MLP tail (L2-resident streaming).
//   logit[e] = ReLU(P[src,0:64] + P[dst,64:128] + b1) . w2 + b2
// One thread per edge; 2 x 16 float4 loads (contiguous 256B per gather).
// b1/W2/b2 are uniform -> scalar loads.
// ---------------------------------------------------------------------------
__global__ __launch_bounds__(256) void edge_mlp(
    const long long* __restrict__ eidx,  // [2, E] int64
    const float* __restrict__ P,         // [N, 128]
    const float* __restrict__ b1,        // [64]
    const float* __restrict__ W2,        // [64]
    const float* __restrict__ b2,        // [1]
    float* __restrict__ out,             // [E]
    int E)
{
    const int e = blockIdx.x * blockDim.x + threadIdx.x;
    if (e >= E) return;

    const long long s = eidx[e];
    const long long d = eidx[(long long)E + e];

    const float4* __restrict__ pa = (const float4*)(P + (size_t)s * 128);
    const float4* __restrict__ pb = (const float4*)(P + (size_t)d * 128 + 64);
    const float4* __restrict__ vb1 = (const float4*)b1;
    const float4* __restrict__ vw2 = (const float4*)W2;

    float acc = b2[0];
#pragma unroll
    for (int i = 0; i < 16; ++i) {
        const float4 a  = pa[i];
        const float4 b  = pb[i];
        const float4 c1 = vb1[i];
        const float4 w  = vw2[i];
        const float h0 = fmaxf(a.x + b.x + c1.x, 0.f);
        const float h1 = fmaxf(a.y + b.y + c1.y, 0.f);
        const float h2 = fmaxf(a.z + b.z + c1.z, 0.f);
        const float h3 = fmaxf(a.w + b.w + c1.w, 0.f);
        acc = fmaf(h0, w.x, acc);
        acc = fmaf(h1, w.y, acc);
        acc = fmaf(h2, w.z, acc);
        acc = fmaf(h3, w.w, acc);
    }
    out[e] = acc;
}

// ---------------------------------------------------------------------------
extern "C" void kernel_launch(void* const* d_in, const int* in_sizes, int n_in,
                              void* d_out, int out_size, void* d_ws, size_t ws_size,
                              hipStream_t stream)
{
    const float*     emb  = (const float*)d_in[0];      // [N, 64] f32
    const long long* eidx = (const long long*)d_in[1];  // [2, E] i64
    const float*     W1   = (const float*)d_in[2];      // [128, 64] f32
    const float*     b1   = (const float*)d_in[3];      // [64] f32
    const float*     W2   = (const float*)d_in[4];      // [64, 1] f32
    const float*     b2   = (const float*)d_in[5];      // [1] f32
    float*           out  = (float*)d_out;              // [E] f32

    const int N = in_sizes[0] / HDIM;       // 50000
    const int E = in_sizes[1] / 2;          // 800000

    float* P = (float*)d_ws;                // needs N*128*4 = 25.6 MB

    // Pass 1: WMMA GEMM  [N x 64] @ [64 x 128] -> P
    node_precompute_wmma<<<N / 16, 256, 0, stream>>>(emb, W1, P, N);

    // Pass 2: per-edge gather + ReLU + dot
    edge_mlp<<<(E + 255) / 256, 256, 0, stream>>>(eidx, P, b1, W2, b2, out, E);
}